// Inferer_37890201485382
// MI455X (gfx1250) — compile-verified
//
#include <hip/hip_runtime.h>
#include <hip/hip_bf16.h>

typedef __attribute__((ext_vector_type(2))) float v2f;
typedef __attribute__((ext_vector_type(8))) float v8f;
typedef int vsi4 __attribute__((vector_size(16)));   // matches builtin param type

#define DIM     256
#define SLOTS   8
#define NMEM    1000000
#define TROWS   16
#define NT16    ((NMEM + TROWS - 1) / TROWS)   // 62500
#define MSTRIDE 260    // padded LDS row stride (bank-conflict-free: 260 % 64 == 4)
#define KSTRIDE 260
#define PW_STRIDE 20   // per-wave P staging row stride
#define PART_STRIDE 2064   // per-block partial: m[8] + l[8] + acc[8*256]
#define MAXB 512

// LDS layout (floats) for attn kernel
#define KLD_OFF   0
#define MBUF_OFF  (16 * KSTRIDE)                 // 4160
#define MBUF_SZ   (TROWS * MSTRIDE)              // 4160 per buffer
#define PW_OFF    (MBUF_OFF + 16 * MBUF_SZ)      // 8 waves x 2 buffers
#define LDS_FLOATS (PW_OFF + 8 * 16 * PW_STRIDE) // 73280 floats = 293120 B (<320KB)

#if defined(__has_builtin)
#if __has_builtin(__builtin_amdgcn_global_load_async_to_lds_b128) && \
    __has_builtin(__builtin_amdgcn_s_wait_asynccnt)
#define HAVE_ASYNC 1
#endif
#endif

#if defined(HAVE_ASYNC)
#define ASYNC_WAIT() __builtin_amdgcn_s_wait_asynccnt(0)
#else
#define ASYNC_WAIT()
#endif

// issue one 16-row tile load (16KB) into this wave's LDS buffer
__device__ __forceinline__ void tile_load(const float* __restrict__ mem,
                                          float* __restrict__ Mb,
                                          size_t row0, int lane) {
#pragma unroll 4
  for (int q = 0; q < 32; ++q) {
    const int lin = q * 32 + lane;        // 16-byte units within the tile
    const int r  = lin >> 6;              // row 0..15
    const int cf = (lin & 63) << 2;       // float column 0..252
    size_t gr = row0 + (size_t)r;
    if (gr >= (size_t)NMEM) gr = 0;       // clamped (masked later)
#if defined(HAVE_ASYNC)
    __builtin_amdgcn_global_load_async_to_lds_b128(
        (__attribute__((address_space(1))) vsi4*)(mem + gr * DIM + cf),
        (__attribute__((address_space(3))) vsi4*)(Mb + r * MSTRIDE + cf), 0, 0);
#else
    const float4 v = *(const float4*)(mem + gr * DIM + cf);
    float* dst = &Mb[r * MSTRIDE + cf];
    dst[0] = v.x; dst[1] = v.y; dst[2] = v.z; dst[3] = v.w;
#endif
  }
}

// ---------------------------------------------------------------------------
// Kernel 1: qspace = tanh([state;heads] @ qW^T + qb); k = heads*qspace
// ---------------------------------------------------------------------------
__global__ void prep_kernel(const float* __restrict__ heads,
                            const float* __restrict__ state,
                            const float* __restrict__ qW,
                            const float* __restrict__ qb,
                            float* __restrict__ kws,
                            float* __restrict__ rs) {
  __shared__ float sh[DIM + SLOTS * DIM];
  const int tid = threadIdx.x;
  sh[tid] = state[tid];
  for (int s = 0; s < SLOTS; ++s) sh[DIM + s * DIM + tid] = heads[s * DIM + tid];
  __syncthreads();

  const float* row = qW + (size_t)tid * (2 * DIM);
  float c0 = 0.f;
  for (int j = 0; j < DIM; ++j) c0 += sh[j] * row[j];
  const float bias = qb[tid];
  for (int s = 0; s < SLOTS; ++s) {
    float a = c0 + bias;
    const float* hs = &sh[DIM + s * DIM];
    for (int j = 0; j < DIM; ++j) a += hs[j] * row[DIM + j];
    const float qs = tanhf(a);
    kws[s * DIM + tid] = hs[tid] * qs;
    rs[s * DIM + tid]  = qs;
  }
  for (int s = SLOTS; s < 16; ++s) kws[s * DIM + tid] = 0.f;
}

// ---------------------------------------------------------------------------
// Kernel 2: wave-autonomous fused streaming softmax-attention pass.
// Each of the 8 waves streams its own double-buffered 16-row tiles (async LDS
// loads), runs GEMM1 + online softmax + GEMM2 entirely in-wave with a private
// 16x256 f32 WMMA accumulator; the 8 wave states merge once at block end.
// ---------------------------------------------------------------------------
__global__ void __launch_bounds__(256)
attn_pass_kernel(const float* __restrict__ mem,
                 const float* __restrict__ kws,
                 float* __restrict__ part,
                 int nblocks) {
  extern __shared__ float lds[];
  float* Kld = lds + KLD_OFF;

  const int tid  = threadIdx.x;
  const int w    = tid >> 5;
  const int lane = tid & 31;
  const int l16  = lane & 15;
  const int hi   = lane >> 4;   // 0: slots 0-7 / K lo, 1: slots 8-15 / K hi

  float* Mbuf0 = lds + MBUF_OFF + (w * 2 + 0) * MBUF_SZ;
  float* Mbuf1 = lds + MBUF_OFF + (w * 2 + 1) * MBUF_SZ;
  float* Pw    = lds + PW_OFF + w * 16 * PW_STRIDE;

  // stage padded Kp (shared) and zero this wave's P staging (pad rows stay 0)
  for (int i = tid; i < 16 * DIM; i += 256) {
    const int r = i >> 8, c = i & 255;
    Kld[r * KSTRIDE + c] = kws[i];
  }
  for (int i = lane; i < 16 * PW_STRIDE; i += 32) Pw[i] = 0.f;

  const size_t me   = (size_t)blockIdx.x * 8 + w;
  const size_t step = (size_t)nblocks * 8;

  v8f acc[16];
#pragma unroll
  for (int j = 0; j < 16; ++j) acc[j] = (v8f){};
  float m_run[8], l_run[8];
#pragma unroll
  for (int i = 0; i < 8; ++i) { m_run[i] = -1e30f; l_run[i] = 0.f; }

  if (me < (size_t)NT16) tile_load(mem, Mbuf0, me * TROWS, lane);
  __syncthreads();   // Kld visible to all waves

  int pbuf = 0;
  for (size_t t = me; t < (size_t)NT16; t += step, pbuf ^= 1) {
    ASYNC_WAIT();   // this wave's current tile resident in LDS
    const size_t tn = t + step;
    float* Mb = pbuf ? Mbuf1 : Mbuf0;
    if (tn < (size_t)NT16) tile_load(mem, pbuf ? Mbuf0 : Mbuf1, tn * TROWS, lane);

    // ---- GEMM1: S[16 slots x 16 rows] = Kp @ Mtile^T (64 WMMAs) ----------
    v8f S = {};
    {
      const float* ka = &Kld[l16 * KSTRIDE + 2 * hi];
      const float* mb = &Mb[l16 * MSTRIDE + 2 * hi];
#pragma unroll 8
      for (int kk = 0; kk < 64; ++kk) {
        const v2f a = *(const v2f*)(ka + 4 * kk);
        const v2f b = *(const v2f*)(mb + 4 * kk);
        S = __builtin_amdgcn_wmma_f32_16x16x4_f32(false, a, false, b,
                                                  (short)0, S, false, false);
      }
    }
    if (t * TROWS + (size_t)l16 >= (size_t)NMEM) {
#pragma unroll
      for (int i = 0; i < 8; ++i) S[i] = -1e30f;
    }

    // ---- batched per-slot max over the 16-lane half ----------------------
    float red[8];
#pragma unroll
    for (int i = 0; i < 8; ++i) red[i] = S[i];
#pragma unroll
    for (int st = 1; st < 16; st <<= 1) {
      float o[8];
#pragma unroll
      for (int i = 0; i < 8; ++i) o[i] = __shfl_xor(red[i], st, 16);
#pragma unroll
      for (int i = 0; i < 8; ++i) red[i] = fmaxf(red[i], o[i]);
    }

    float corr[8], p[8];
#pragma unroll
    for (int i = 0; i < 8; ++i) {
      const float nm = fmaxf(m_run[i], red[i]);
      corr[i] = __expf(m_run[i] - nm);
      m_run[i] = nm;
      p[i] = hi ? 0.f : __expf(S[i] - nm);
      if (!hi) Pw[i * PW_STRIDE + l16] = p[i];   // transpose stage for GEMM2 A
    }

    // ---- batched per-slot sum of P ---------------------------------------
#pragma unroll
    for (int st = 1; st < 16; st <<= 1) {
      float o[8];
#pragma unroll
      for (int i = 0; i < 8; ++i) o[i] = __shfl_xor(p[i], st, 16);
#pragma unroll
      for (int i = 0; i < 8; ++i) p[i] += o[i];
    }
#pragma unroll
    for (int i = 0; i < 8; ++i) l_run[i] = l_run[i] * corr[i] + p[i];

    // ---- rescale accumulators (padded slots stay zero) -------------------
    float csc[8];
#pragma unroll
    for (int i = 0; i < 8; ++i) csc[i] = hi ? 1.f : corr[i];
#pragma unroll
    for (int j = 0; j < 16; ++j)
#pragma unroll
      for (int i = 0; i < 8; ++i) acc[j][i] *= csc[i];

    // ---- GEMM2: acc[16x256] += P[16x16] @ Mtile[16x256] (64 WMMAs) -------
    v2f A2[4];
#pragma unroll
    for (int kk = 0; kk < 4; ++kk)
      A2[kk] = *(const v2f*)(Pw + l16 * PW_STRIDE + 4 * kk + 2 * hi);
#pragma unroll
    for (int j = 0; j < 16; ++j) {
      const float* m0 = &Mb[(2 * hi) * MSTRIDE + j * 16 + l16];
#pragma unroll
      for (int kk = 0; kk < 4; ++kk) {
        const float* r0 = m0 + kk * 4 * MSTRIDE;
        v2f b; b.x = r0[0]; b.y = r0[MSTRIDE];
        acc[j] = __builtin_amdgcn_wmma_f32_16x16x4_f32(false, A2[kk], false, b,
                                                       (short)0, acc[j], false, false);
      }
    }
  }

  // ---- merge the 8 wave states within the block --------------------------
  __syncthreads();                       // all waves done with their buffers
  float* Cmb = lds + MBUF_OFF;           // reuse tile-buffer region
  if (lane == 0) {
#pragma unroll
    for (int i = 0; i < 8; ++i) {
      Cmb[w * PART_STRIDE + i]     = m_run[i];
      Cmb[w * PART_STRIDE + 8 + i] = l_run[i];
    }
  }
  if (!hi) {
#pragma unroll
    for (int j = 0; j < 16; ++j)
#pragma unroll
      for (int i = 0; i < 8; ++i)
        Cmb[w * PART_STRIDE + 16 + i * DIM + j * 16 + l16] = acc[j][i];
  }
  __syncthreads();

  float* pb = part + (size_t)blockIdx.x * PART_STRIDE;
  const int d = tid;   // 256 threads: one dim each
  for (int s = 0; s < SLOTS; ++s) {
    float gm = Cmb[s];
    for (int w2 = 1; w2 < 8; ++w2) gm = fmaxf(gm, Cmb[w2 * PART_STRIDE + s]);
    float gl = 0.f, rv = 0.f;
    for (int w2 = 0; w2 < 8; ++w2) {
      const float sc = __expf(Cmb[w2 * PART_STRIDE + s] - gm);
      gl += Cmb[w2 * PART_STRIDE + 8 + s] * sc;
      rv += Cmb[w2 * PART_STRIDE + 16 + s * DIM + d] * sc;
    }
    pb[16 + s * DIM + d] = rv;
    if (tid == s) { pb[s] = gm; pb[8 + s] = gl; }
  }
}

// ---------------------------------------------------------------------------
// Kernel 3: merge block partials -> raw -> new_heads -> new_state -> reveal
// ---------------------------------------------------------------------------
__global__ void finalize_kernel(const float* __restrict__ part, int nb,
                                const float* __restrict__ heads,
                                const float* __restrict__ state,
                                const float* __restrict__ sW,
                                const float* __restrict__ sb,
                                const float* __restrict__ rvW,
                                const float* __restrict__ rvb,
                                const float* __restrict__ rs,
                                float* __restrict__ out) {
  __shared__ float gm[8], gl[8];
  __shared__ float scale[MAXB * 8];
  __shared__ float cat[DIM * (SLOTS + 1)];
  __shared__ float red[256];
  __shared__ float rev;
  const int tid = threadIdx.x;

  if (tid < 8) {
    const int s = tid;
    float m = -1e30f;
    for (int b = 0; b < nb; ++b) m = fmaxf(m, part[(size_t)b * PART_STRIDE + s]);
    float l = 0.f;
    for (int b = 0; b < nb; ++b)
      l += part[(size_t)b * PART_STRIDE + 8 + s] *
           __expf(part[(size_t)b * PART_STRIDE + s] - m);
    gm[s] = m;
    gl[s] = (l > 0.f) ? l : 1.f;
  }
  __syncthreads();

  for (int i = tid; i < nb * 8; i += 256) {
    const int b = i >> 3, s = i & 7;
    scale[i] = __expf(part[(size_t)b * PART_STRIDE + s] - gm[s]) / gl[s];
  }
  cat[tid] = state[tid];
  __syncthreads();

  for (int s = 0; s < SLOTS; ++s) {
    float r = 0.f;
    for (int b = 0; b < nb; ++b)
      r += part[(size_t)b * PART_STRIDE + 16 + s * DIM + tid] * scale[b * 8 + s];
    const float rsp = rs[s * DIM + tid];
    const float h   = heads[s * DIM + tid];
    cat[DIM + s * DIM + tid] = r * rsp + (1.f - rsp) * h;
  }
  __syncthreads();

  const float* wrow = sW + (size_t)tid * (DIM * (SLOTS + 1));
  float a = sb[tid];
  for (int j = 0; j < DIM * (SLOTS + 1); ++j) a += cat[j] * wrow[j];

  red[tid] = a * rvW[tid];
  __syncthreads();
  for (int off = 128; off > 0; off >>= 1) {
    if (tid < off) red[tid] += red[tid + off];
    __syncthreads();
  }
  if (tid == 0) rev = 1.f / (1.f + __expf(-(red[0] + rvb[0])));
  __syncthreads();

  out[tid] = a * rev;
}

// ---------------------------------------------------------------------------
extern "C" void kernel_launch(void* const* d_in, const int* in_sizes, int n_in,
                              void* d_out, int out_size, void* d_ws, size_t ws_size,
                              hipStream_t stream) {
  (void)in_sizes; (void)n_in; (void)out_size;
  const float* mem   = (const float*)d_in[0];
  const float* heads = (const float*)d_in[1];
  const float* state = (const float*)d_in[2];
  const float* qW    = (const float*)d_in[3];
  const float* qb    = (const float*)d_in[4];
  const float* sW    = (const float*)d_in[5];
  const float* sb    = (const float*)d_in[6];
  const float* rvW   = (const float*)d_in[7];
  const float* rvb   = (const float*)d_in[8];

  float* ws   = (float*)d_ws;
  float* kws  = ws;          // Kp: 16*256
  float* rs   = ws + 4096;   // rspace: 8*256
  float* part = ws + 6144;   // per-block partials

  const size_t avail = ws_size / sizeof(float);
  long maxb = ((long)avail - 6144) / PART_STRIDE;
  int nb = (int)(maxb < 1 ? 1 : (maxb > MAXB ? MAXB : maxb));

  prep_kernel<<<1, 256, 0, stream>>>(heads, state, qW, qb, kws, rs);

  attn_pass_kernel<<<nb, 256, (size_t)LDS_FLOATS * sizeof(float), stream>>>(
      mem, kws, part, nb);

  finalize_kernel<<<1, 256, 0, stream>>>(part, nb, heads, state, sW, sb, rvW, rvb,
                                         rs, (float*)d_out);
}